// SelfAttention_68204080661140
// MI455X (gfx1250) — compile-verified
//
#include <hip/hip_runtime.h>
#include <math.h>

// MI455X / gfx1250, wave32. fp32 self-attention block using V_WMMA_F32_16X16X4_F32
// + async global->LDS staging (ASYNCcnt), double-buffered K/V tiles, and
// WMMA-operand-friendly intermediate layouts (Q/K stored (n,d); V stored (d,n))
// so every A/B fragment pair is a single contiguous b64 load.

typedef float v2f __attribute__((ext_vector_type(2)));
typedef float v8f __attribute__((ext_vector_type(8)));
typedef int   v4i __attribute__((vector_size(16)));   // matches builtin param type

#define C_IN   256
#define C_QKV  768
#define HW     4096
#define HEADS  4
#define HD     64
#define KPAD   68      // K-tile LDS row stride: 272B = 16B-aligned, bank stride 4
#define BPAD   260     // GEMM X-tile LDS row stride: 8B-aligned, bank stride 4

typedef __attribute__((address_space(1))) v4i gv4i;   // global int4
typedef __attribute__((address_space(3))) v4i lv4i;   // LDS int4

#if defined(__has_builtin)
#if __has_builtin(__builtin_amdgcn_global_load_async_to_lds_b128)
#define HAVE_ASYNC_COPY 1
#endif
#if __has_builtin(__builtin_amdgcn_s_wait_asynccnt)
#define WAIT_ASYNC(n) __builtin_amdgcn_s_wait_asynccnt(n)
#endif
#endif
#ifndef WAIT_ASYNC
#define WAIT_ASYNC(n) asm volatile("s_wait_asynccnt %0" ::"i"(n) : "memory")
#endif

__device__ __forceinline__ v8f wmma_f32(v2f a, v2f b, v8f c) {
  // (neg_a, A, neg_b, B, c_mod, C, reuse_a, reuse_b)
  return __builtin_amdgcn_wmma_f32_16x16x4_f32(false, a, false, b, (short)0, c,
                                               false, false);
}

__device__ __forceinline__ void async_cp16(const float* src, float* dst) {
#ifdef HAVE_ASYNC_COPY
  __builtin_amdgcn_global_load_async_to_lds_b128((gv4i*)src, (lv4i*)dst, 0, 0);
#else
  *(float4*)dst = *(const float4*)src;
#endif
}

// ---------------------------------------------------------------------------
// GEMM: acc[m,n] = sum_k W[m,k] * X[bb,k,n] + bias[m].
// QKV=false: Y[bb,m,n] = acc (+ resid). QKV=true: scatter into qT/kT ((b,h,n,d))
// and vB ((b,h,d,n)) so the attention kernel gets contiguous operand pairs.
// K fixed at 256. One 16x16 tile per wave; X tile staged transposed in LDS.
// grid: (M/128, HW/16, batch), block: 256 threads (8 waves).
// ---------------------------------------------------------------------------
template <bool QKV, bool RES>
__global__ __launch_bounds__(256)
void gemm_bias_kernel(const float* __restrict__ W, const float* __restrict__ X,
                      const float* __restrict__ bias,
                      const float* __restrict__ resid,
                      float* __restrict__ Y,
                      float* __restrict__ qT, float* __restrict__ kT,
                      float* __restrict__ vB, int M) {
  __shared__ float ldsBt[16 * BPAD];   // X tile transposed: [n][k], padded rows

  const int tid  = threadIdx.x;
  const int lane = tid & 31;
  const int wid  = tid >> 5;
  const int hi   = (lane >> 4) & 1;  // lanes 16-31
  const int l16  = lane & 15;
  const int kb   = hi * 2;           // K offset for hi half (ISA A/B layout)
  const int bb   = blockIdx.z;
  const int nstart = blockIdx.y * 16;
  const int mtile  = blockIdx.x * 8 + wid;

  const float* Xb = X + (size_t)bb * C_IN * HW;

  // Cooperative transposed load: 16 consecutive threads load 16 consecutive n
  {
    const int n_l  = tid & 15;
    const int kgrp = tid >> 4;       // 0..15
    for (int i = 0; i < 16; ++i) {
      const int k = kgrp + i * 16;
      ldsBt[n_l * BPAD + k] = Xb[(size_t)k * HW + nstart + n_l];
    }
  }
  __syncthreads();

  const int    mrow = mtile * 16 + l16;
  const float* Wr   = W + (size_t)mrow * C_IN;
  const float* Brow = ldsBt + l16 * BPAD;

  v8f acc = {0.f, 0.f, 0.f, 0.f, 0.f, 0.f, 0.f, 0.f};
  for (int kk = 0; kk < 64; ++kk) {        // K = 256 = 64 steps of 4
    const int k0 = kk * 4 + kb;
    const v2f a = *(const v2f*)(Wr + k0);     // global b64
    const v2f b = *(const v2f*)(Brow + k0);   // ds b64, conflict-free
    acc = wmma_f32(a, b, acc);
  }

  if (QKV) {
    // Tile-uniform head / q-k-v class (192 and 64 are multiples of 16)
    const int m0  = mtile * 16;
    const int h   = m0 / 192;
    const int cls = (m0 % 192) / 64;
    const int n   = nstart + l16;
    for (int r = 0; r < 8; ++r) {
      const int   m  = m0 + r + 8 * hi;       // D layout: rows r / r+8
      const int   dd = m & 63;
      const float v  = acc[r] + bias[m];
      if (cls == 0)
        qT[((size_t)(bb * HEADS + h) * HW + n) * HD + dd] = v;
      else if (cls == 1)
        kT[((size_t)(bb * HEADS + h) * HW + n) * HD + dd] = v;
      else
        vB[((size_t)(bb * HEADS + h) * HD + dd) * HW + n] = v;
    }
  } else {
    float*       Yb = Y + (size_t)bb * M * HW;
    const float* Rb = resid + (size_t)bb * M * HW;
    for (int r = 0; r < 8; ++r) {
      const int    m   = mtile * 16 + r + 8 * hi;
      const size_t off = (size_t)m * HW + nstart + l16;
      float v = acc[r] + bias[m];
      if (RES) v += Rb[off];
      Yb[off] = v;
    }
  }
}

// ---------------------------------------------------------------------------
// Flash attention per (b, head), reading qT/kT ((n,d)) and vB ((d,n)).
// Output written with the reference reshape: out[bb][h*HW*HD + n*HD + d].
// grid: (HW/128, HEADS, batch), block: 256 (8 waves; 16 query rows per wave).
// Double-buffered LDS tiles streamed by async DMA; one online-softmax update
// per 64-key tile (S held as 4 D-blocks in VGPRs).
// ---------------------------------------------------------------------------
__global__ __launch_bounds__(256)
void attn_kernel(const float* __restrict__ qT, const float* __restrict__ kT,
                 const float* __restrict__ vB, float* __restrict__ out) {
  __shared__ float ldsK[2 * 64 * KPAD];  // K tile: [m][d], padded rows
  __shared__ float ldsV[2 * 64 * 64];    // V tile: [d][m]
  __shared__ float ldsP[8 * 256];        // per-wave P transpose scratch

  const int tid  = threadIdx.x;
  const int lane = tid & 31;
  const int wid  = tid >> 5;
  const int hi   = (lane >> 4) & 1;
  const int l16  = lane & 15;
  const int kb   = hi * 2;
  const int qblk = blockIdx.x;       // 0..31
  const int head = blockIdx.y;       // 0..3
  const int bb   = blockIdx.z;       // 0..1

  const float* Qt = qT + (size_t)(bb * HEADS + head) * HW * HD;  // [n][d]
  const float* Kt = kT + (size_t)(bb * HEADS + head) * HW * HD;  // [n][d]
  const float* Vb = vB + (size_t)(bb * HEADS + head) * HD * HW;  // [d][n]

  const int nq = qblk * 128 + wid * 16 + l16;   // this lane's query row

  // Q fragments: A[m=n_q][k=d] pairs are contiguous in (n,d) layout -> b64.
  // Fold the 1/sqrt(64) softmax scale in here.
  v2f qf[16];
  for (int kd = 0; kd < 16; ++kd) {
    const v2f q = *(const v2f*)(Qt + (size_t)nq * HD + kd * 4 + kb);
    qf[kd].x = q.x * 0.125f;
    qf[kd].y = q.y * 0.125f;
  }

  v8f oacc[4];
  for (int db = 0; db < 4; ++db)
    oacc[db] = (v8f){0.f, 0.f, 0.f, 0.f, 0.f, 0.f, 0.f, 0.f};
  float mrun[8], lrun[8];
  for (int r = 0; r < 8; ++r) { mrun[r] = -INFINITY; lrun[r] = 0.f; }

  float* myP = ldsP + wid * 256;

  // Stage one 64-key tile: K is a contiguous 16KB block in (n,d) layout;
  // V rows are (d, mstart..mstart+63). 8 b128 async ops per thread.
  auto stage = [&](float* dK, float* dV, int mstart) {
    const float* srcK = Kt + (size_t)mstart * HD;
    for (int i = 0; i < 4; ++i) {
      const int id = tid + i * 256;    // 0..1023 float4 slots
      const int m  = id >> 4;
      const int j  = id & 15;
      async_cp16(srcK + id * 4, dK + m * KPAD + j * 4);
      async_cp16(Vb + (size_t)m * HW + mstart + j * 4, dV + m * 64 + j * 4);
    }
  };

  // Prologue: stage tile 0 into buffer 0
  stage(ldsK, ldsV, 0);

  for (int it = 0; it < 64; ++it) {            // stream 64 keys per iteration
    const int cur = it & 1;
    float* bKt = ldsK + cur * (64 * KPAD);
    float* bV  = ldsV + cur * 4096;

    // Kick off the next tile into the other buffer (overlaps compute below)
    if (it + 1 < 64)
      stage(ldsK + (cur ^ 1) * (64 * KPAD), ldsV + (cur ^ 1) * 4096,
            (it + 1) * 64);
#ifdef HAVE_ASYNC_COPY
    if (it + 1 < 64) { WAIT_ASYNC(8); }        // tile 'it' (my share) done
    else             { WAIT_ASYNC(0); }
#endif
    __syncthreads();                           // all shares visible

    // ---- All four S blocks of this 64-key tile (pre-scaled by Q) ----
    v8f S[4];
    for (int sub = 0; sub < 4; ++sub) {
      const float* rowK = bKt + (sub * 16 + l16) * KPAD;  // this lane's key row
      v8f s = {0.f, 0.f, 0.f, 0.f, 0.f, 0.f, 0.f, 0.f};
      for (int kd = 0; kd < 16; ++kd) {
        const v2f b = *(const v2f*)(rowK + kd * 4 + kb);  // single ds b64
        s = wmma_f32(qf[kd], b, s);
      }
      S[sub] = s;
    }

    // ---- One joint online-softmax update for all 64 keys ----
    float alpha[8];
    for (int r = 0; r < 8; ++r) {
      float rm = fmaxf(fmaxf(S[0][r], S[1][r]), fmaxf(S[2][r], S[3][r]));
      rm = fmaxf(rm, __shfl_xor(rm, 1, 32));
      rm = fmaxf(rm, __shfl_xor(rm, 2, 32));
      rm = fmaxf(rm, __shfl_xor(rm, 4, 32));
      rm = fmaxf(rm, __shfl_xor(rm, 8, 32));
      const float mnew = fmaxf(mrun[r], rm);
      alpha[r] = __expf(mrun[r] - mnew);
      mrun[r]  = mnew;
      const float p0 = __expf(S[0][r] - mnew);
      const float p1 = __expf(S[1][r] - mnew);
      const float p2 = __expf(S[2][r] - mnew);
      const float p3 = __expf(S[3][r] - mnew);
      S[0][r] = p0; S[1][r] = p1; S[2][r] = p2; S[3][r] = p3;
      float rs = (p0 + p1) + (p2 + p3);
      rs += __shfl_xor(rs, 1, 32);
      rs += __shfl_xor(rs, 2, 32);
      rs += __shfl_xor(rs, 4, 32);
      rs += __shfl_xor(rs, 8, 32);
      lrun[r] = lrun[r] * alpha[r] + rs;
    }
    // Rescale O accumulators once per tile
    for (int db = 0; db < 4; ++db)
      for (int r = 0; r < 8; ++r)
        oacc[db][r] *= alpha[r];

    // ---- O(16q x 64d) += P(16q x 64k) * V^T(64k x 64d), sub-block at a time
    for (int sub = 0; sub < 4; ++sub) {
      const int mb = sub * 16;
      for (int r = 0; r < 8; ++r)
        myP[(r + 8 * hi) * 16 + l16] = S[sub][r];   // stage P for transpose
      __builtin_amdgcn_wave_barrier();              // DS write -> DS read order
      for (int kk = 0; kk < 4; ++kk) {
        const v2f a = *(const v2f*)(myP + l16 * 16 + kk * 4 + kb);
        for (int db = 0; db < 4; ++db) {
          const float* vrow = bV + (db * 16 + l16) * 64 + mb + kk * 4 + kb;
          const v2f b = *(const v2f*)vrow;          // contiguous along keys
          oacc[db] = wmma_f32(a, b, oacc[db]);
        }
      }
      __builtin_amdgcn_wave_barrier();              // reads before next write
    }
    __syncthreads();                     // tile reads done before overwrite
  }

  // Normalize and store with the reference's direct-reshape layout
  float* Ob = out + (size_t)bb * (C_IN * HW) + (size_t)head * (HW * HD);
  for (int db = 0; db < 4; ++db) {
    for (int r = 0; r < 8; ++r) {
      const int n = qblk * 128 + wid * 16 + r + 8 * hi;
      const int d = db * 16 + l16;
      Ob[(size_t)n * HD + d] = oacc[db][r] / lrun[r];
    }
  }
}

// ---------------------------------------------------------------------------
extern "C" void kernel_launch(void* const* d_in, const int* in_sizes, int n_in,
                              void* d_out, int out_size, void* d_ws,
                              size_t ws_size, hipStream_t stream) {
  const float* x     = (const float*)d_in[0];   // (2,256,64,64)
  const float* w_qkv = (const float*)d_in[1];   // (768,256)
  const float* b_qkv = (const float*)d_in[2];   // (768,)
  const float* w_out = (const float*)d_in[3];   // (256,256)
  const float* b_out = (const float*)d_in[4];   // (256,)
  float* out = (float*)d_out;                   // (2,256,64,64)

  const size_t perhd = (size_t)2 * HEADS * HW * HD;   // 2,097,152 floats
  float* qT   = (float*)d_ws;                         // (b,h,n,d)
  float* kT   = qT + perhd;                           // (b,h,n,d)
  float* vB   = kT + perhd;                           // (b,h,d,n)
  float* attn = vB + perhd;                           // (b, c, hw) reshape

  // 1) QKV projection, scattered into attention-friendly layouts
  gemm_bias_kernel<true, false>
      <<<dim3(C_QKV / 128, HW / 16, 2), 256, 0, stream>>>(
          w_qkv, x, b_qkv, nullptr, nullptr, qT, kT, vB, C_QKV);

  // 2) Flash attention per (b, head)
  attn_kernel<<<dim3(HW / 128, HEADS, 2), 256, 0, stream>>>(qT, kT, vB, attn);

  // 3) Output projection + bias + residual
  gemm_bias_kernel<false, true>
      <<<dim3(C_IN / 128, HW / 16, 2), 256, 0, stream>>>(
          w_out, attn, b_out, x, out, nullptr, nullptr, nullptr, C_IN);
}